// HungarianMatcher_48163763258165
// MI455X (gfx1250) — compile-verified
//
#include <hip/hip_runtime.h>
#include <hip/hip_bf16.h>
#include <stdint.h>

// Problem constants (from reference setup_inputs)
constexpr int NB  = 16;
constexpr int NQ  = 900;
constexpr int NC  = 92;
constexpr int NT  = 1600;
constexpr int BQ  = NB * NQ;      // 14400 rows
constexpr int PST = 96;           // padded prob row stride (16B-multiple)
constexpr int TM  = 64;           // rows per tile
constexpr int TN  = 64;           // cols per tile
constexpr int NRT = BQ / TM;      // 225 row tiles
constexpr int GRID_Y = 45;        // row-tile stride (225/45 = 5 tiles per block)

// ---------------- CDNA5 async global->LDS helpers ----------------
// gfx1250: GLOBAL_LOAD_ASYNC_TO_LDS_B64/B128, tracked by ASYNCcnt.
typedef int v4i __attribute__((ext_vector_type(4)));
typedef int v2i __attribute__((ext_vector_type(2)));
typedef __attribute__((address_space(1))) v4i* g_v4p;
typedef __attribute__((address_space(3))) v4i* l_v4p;
typedef __attribute__((address_space(1))) v2i* g_v2p;
typedef __attribute__((address_space(3))) v2i* l_v2p;

__device__ __forceinline__ void cp_async_b128(const float* g, const float* l) {
  __builtin_amdgcn_global_load_async_to_lds_b128(
      (g_v4p)(uintptr_t)g, (l_v4p)(uintptr_t)l, 0, 0);
}
__device__ __forceinline__ void cp_async_b64(const float* g, const float* l) {
  __builtin_amdgcn_global_load_async_to_lds_b64(
      (g_v2p)(uintptr_t)g, (l_v2p)(uintptr_t)l, 0, 0);
}
__device__ __forceinline__ void wait_async0() {
  asm volatile("s_wait_asynccnt 0" ::: "memory");
}

// ---------------- Kernel: cost matrix (placed first so --disasm snippet shows it) ----
// 64x64 output tile per block iteration; prob/qbox row tiles double-buffered in LDS
// via the gfx1250 async global->LDS engine, overlapping the copy with compute.
__global__ __launch_bounds__(256)
void hm_cost(const float* __restrict__ prob, const float* __restrict__ qdat,
             const float* __restrict__ tdat, float* __restrict__ out) {
  __shared__ __align__(16) float sProb[2][TM * PST];  // 2 x 24 KB
  __shared__ __align__(16) float sQ[2][TM * 8];       // 2 x 2 KB
  __shared__ __align__(16) float sT[TN * 8];          // 2 KB

  const int t    = threadIdx.x;
  const int lane = t & 31;
  const int wid  = t >> 5;
  const int colBase = blockIdx.x * TN;
  const int rtStep  = gridDim.y;

  // Stage target tile once (1 async b64 per thread: 64*8 floats)
  cp_async_b64(tdat + (size_t)colBase * 8 + t * 2, &sT[t * 2]);

  // Stage first row tile into buffer 0
  int rt0 = blockIdx.y;
  {
    const float* gp = prob + (size_t)rt0 * TM * PST;
#pragma unroll
    for (int k = 0; k < 6; ++k) {
      int c4 = (t + 256 * k) * 4;
      cp_async_b128(gp + c4, &sProb[0][c4]);
    }
    cp_async_b64(qdat + (size_t)rt0 * TM * 8 + t * 2, &sQ[0][t * 2]);
  }
  wait_async0();
  __syncthreads();

  // Per-thread column registers (2 columns, fixed for the whole block)
  float tcx[2], tcy[2], tw[2], th[2], tx0[2], ty0[2], tx1[2], ty1[2], tarea[2];
  int   tid_[2];
#pragma unroll
  for (int kc = 0; kc < 2; ++kc) {
    const float* p = &sT[(lane + 32 * kc) * 8];
    tcx[kc] = p[0]; tcy[kc] = p[1]; tw[kc] = p[2]; th[kc] = p[3];
    tid_[kc] = __float_as_int(p[4]);
    tx0[kc] = tcx[kc] - 0.5f * tw[kc];
    ty0[kc] = tcy[kc] - 0.5f * th[kc];
    tx1[kc] = tcx[kc] + 0.5f * tw[kc];
    ty1[kc] = tcy[kc] + 0.5f * th[kc];
    tarea[kc] = tw[kc] * th[kc];
  }

  int buf = 0;
  for (int rt = rt0; rt < NRT; rt += rtStep) {
    int rtn = rt + rtStep;
    bool hasNext = rtn < NRT;
    if (hasNext) {  // issue async stage of next tile into other buffer
      const float* gp = prob + (size_t)rtn * TM * PST;
#pragma unroll
      for (int k = 0; k < 6; ++k) {
        int c4 = (t + 256 * k) * 4;
        cp_async_b128(gp + c4, &sProb[buf ^ 1][c4]);
      }
      cp_async_b64(qdat + (size_t)rtn * TM * 8 + t * 2, &sQ[buf ^ 1][t * 2]);
    }

    const float* sp = sProb[buf];
    const float* sq = sQ[buf];
    const int rowBase = rt * TM;
#pragma unroll
    for (int kr = 0; kr < 8; ++kr) {
      int rl = wid + 8 * kr;
      const float* q = &sq[rl * 8];
      float qcx = q[0], qcy = q[1], qw = q[2], qh = q[3];
      float qx0 = q[4], qy0 = q[5], qx1 = q[6], qy1 = q[7];
      float qarea = qw * qh;
      const float* pr = &sp[rl * PST];
      float* orow = out + (size_t)(rowBase + rl) * NT + colBase;
#pragma unroll
      for (int kc = 0; kc < 2; ++kc) {
        float l1 = fabsf(qcx - tcx[kc]) + fabsf(qcy - tcy[kc]) +
                   fabsf(qw - tw[kc]) + fabsf(qh - th[kc]);
        float ix0 = fmaxf(qx0, tx0[kc]), iy0 = fmaxf(qy0, ty0[kc]);
        float ix1 = fminf(qx1, tx1[kc]), iy1 = fminf(qy1, ty1[kc]);
        float iw = fmaxf(ix1 - ix0, 0.f), ih = fmaxf(iy1 - iy0, 0.f);
        float inter = iw * ih;
        float uni = qarea + tarea[kc] - inter;
        // enclosing box: boxes are well-formed (w,h>=0) so ew,eh >= 0; no clamp needed
        float ew = fmaxf(qx1, tx1[kc]) - fminf(qx0, tx0[kc]);
        float eh = fmaxf(qy1, ty1[kc]) - fminf(qy0, ty0[kc]);
        float earea = ew * eh;
        float p = pr[tid_[kc]];
        // giou = iou - (area_e-union)/area_e = inter/uni - 1 + uni/earea
        float giou = inter * __builtin_amdgcn_rcpf(uni) - 1.0f +
                     uni * __builtin_amdgcn_rcpf(earea);
        orow[lane + 32 * kc] = 5.0f * l1 - p - 2.0f * giou;
      }
    }
    // Single barrier handles both hazards: (a) all waves finished reading `buf`
    // before next iteration's async issue overwrites it, (b) every wave's async
    // stage of `buf^1` has landed (each wave waits its own ASYNCcnt first).
    wait_async0();
    __syncthreads();
    buf ^= 1;
  }
}

// ---------------- Kernel: softmax + query-box precompute ----------------
// One wave32 per row. prob row padded to 96 floats. qdat: {cx,cy,w,h,x0,y0,x1,y1}
__global__ __launch_bounds__(256)
void hm_prep_queries(const float* __restrict__ logits, const float* __restrict__ boxes,
                     float* __restrict__ prob, float* __restrict__ qdat) {
  int gw   = blockIdx.x * 8 + (threadIdx.x >> 5);
  int lane = threadIdx.x & 31;
  if (gw >= BQ) return;
  const float* L = logits + (size_t)gw * NC;
  float a0 = L[lane];
  float a1 = L[lane + 32];
  bool  has2 = lane < (NC - 64);           // lanes 0..27
  float a2 = has2 ? L[lane + 64] : -3.0e38f;
  float m = fmaxf(fmaxf(a0, a1), a2);
#pragma unroll
  for (int o = 16; o > 0; o >>= 1) m = fmaxf(m, __shfl_xor(m, o, 32));
  float e0 = __expf(a0 - m), e1 = __expf(a1 - m);
  float e2 = has2 ? __expf(a2 - m) : 0.0f;
  float s = e0 + e1 + e2;
#pragma unroll
  for (int o = 16; o > 0; o >>= 1) s += __shfl_xor(s, o, 32);
  float inv = __builtin_amdgcn_rcpf(s);
  float* P = prob + (size_t)gw * PST;
  P[lane]      = e0 * inv;
  P[lane + 32] = e1 * inv;
  P[lane + 64] = has2 ? e2 * inv : 0.0f;   // covers pad 92..95 too
  if (lane == 0) {
    float4 b = *reinterpret_cast<const float4*>(boxes + (size_t)gw * 4);
    float x0 = b.x - 0.5f * b.z, y0 = b.y - 0.5f * b.w;
    float x1 = b.x + 0.5f * b.z, y1 = b.y + 0.5f * b.w;
    float* qd = qdat + (size_t)gw * 8;
    *reinterpret_cast<float4*>(qd)     = make_float4(b.x, b.y, b.z, b.w);
    *reinterpret_cast<float4*>(qd + 4) = make_float4(x0, y0, x1, y1);
  }
}

// ---------------- Kernel: target precompute ----------------
// tdat: {cx,cy,w,h, bitcast(id), 0,0,0}
__global__ __launch_bounds__(256)
void hm_prep_targets(const int* __restrict__ ids, const float* __restrict__ boxes,
                     float* __restrict__ tdat) {
  int i = blockIdx.x * blockDim.x + threadIdx.x;
  if (i >= NT) return;
  float4 b = *reinterpret_cast<const float4*>(boxes + (size_t)i * 4);
  float* td = tdat + (size_t)i * 8;
  *reinterpret_cast<float4*>(td)     = make_float4(b.x, b.y, b.z, b.w);
  *reinterpret_cast<float4*>(td + 4) = make_float4(__int_as_float(ids[i]), 0.f, 0.f, 0.f);
}

extern "C" void kernel_launch(void* const* d_in, const int* in_sizes, int n_in,
                              void* d_out, int out_size, void* d_ws, size_t ws_size,
                              hipStream_t stream) {
  const float* logits = (const float*)d_in[0];   // [16,900,92]
  const float* qboxes = (const float*)d_in[1];   // [16,900,4]
  const int*   ids    = (const int*)d_in[2];     // [1600]
  const float* tboxes = (const float*)d_in[3];   // [1600,4]
  float* out = (float*)d_out;                    // [16,900,1600]

  float* ws   = (float*)d_ws;
  float* prob = ws;                      // BQ * 96 floats
  float* qdat = prob + (size_t)BQ * PST; // BQ * 8 floats
  float* tdat = qdat + (size_t)BQ * 8;   // NT * 8 floats

  hm_prep_queries<<<BQ / 8, 256, 0, stream>>>(logits, qboxes, prob, qdat);
  hm_prep_targets<<<(NT + 255) / 256, 256, 0, stream>>>(ids, tboxes, tdat);
  dim3 grid(NT / TN, GRID_Y);            // (25, 45)
  hm_cost<<<grid, 256, 0, stream>>>(prob, qdat, tdat, out);
}